// SS2D_57251914055732
// MI455X (gfx1250) — compile-verified
//
#include <hip/hip_runtime.h>
#include <stdint.h>

// Problem shape (fixed by the reference)
#define BSZ 4
#define LSZ 4096
#define DMSZ 1024
#define CPB 32           // channels per block == lanes (wave32)
#define TC 128           // timesteps per chunk
#define NCHUNK (LSZ / TC)

typedef unsigned int v4u __attribute__((ext_vector_type(4)));
typedef int          v4i __attribute__((ext_vector_type(4)));
typedef int          v8i __attribute__((ext_vector_type(8)));

// Issue one TDM 2D tile load: TC rows x CPB floats, row stride DMSZ elements,
// global -> LDS. Tracked by TENSORcnt.
__device__ __forceinline__ void tdm_load_tile(uint32_t lds_byte_addr,
                                              const float* gptr) {
  uint64_t ga = (uint64_t)(uintptr_t)gptr;
  // D# group 0: count=1 | lds_addr | global_addr(57b) | type=2 (bits 127:126)
  v4u g0 = { 1u,
             lds_byte_addr,
             (uint32_t)ga,
             (uint32_t)(ga >> 32) | 0x80000000u };
  // D# group 1:
  //  dw0: data_size=2 (4 bytes) in bits[17:16]; workgroup_mask=0 (not in cluster)
  //  dw1[31:16] = tensor_dim0 low16 = CPB
  //  dw2[31:16] = tensor_dim1 low16 = TC
  //  dw3[31:16] = tile_dim0 = CPB
  //  dw4[15:0]  = tile_dim1 = TC  (tile_dim2 = 0 -> 2D)
  //  dw5        = tensor_dim0_stride low32 = DMSZ (elements)
  v8i g1 = { 0x00020000,
             (CPB << 16),
             (TC  << 16),
             (CPB << 16),
             TC,
             DMSZ,
             0, 0 };
  v4i z4 = { 0, 0, 0, 0 };
#if __has_include(<hip/amd_detail/amd_gfx1250_TDM.h>)
  // therock-10.0 headers present -> clang-23 6-arg builtin
  v8i z8 = { 0, 0, 0, 0, 0, 0, 0, 0 };
  __builtin_amdgcn_tensor_load_to_lds(g0, g1, z4, z4, z8, 0);
#else
  // ROCm 7.2 clang-22 5-arg builtin
  __builtin_amdgcn_tensor_load_to_lds(g0, g1, z4, z4, 0);
#endif
}

__device__ __forceinline__ float softplus_fast(float x) {
  float sp = __logf(1.0f + __expf(x));
  return (x > 20.0f) ? x : sp;
}

__device__ __forceinline__ float tanh_fast(float z) {
#if __has_builtin(__builtin_amdgcn_tanhf)
  return __builtin_amdgcn_tanhf(z);           // v_tanh_f32 (gfx1250 TRANS op)
#elif __has_builtin(__builtin_amdgcn_tanh_f32)
  return __builtin_amdgcn_tanh_f32(z);
#else
  // exp-based: correct saturation at +/-1 for large |z|
  return 1.0f - 2.0f / (__expf(2.0f * z) + 1.0f);
#endif
}

__global__ __launch_bounds__(CPB) void SS2D_scan_kernel(
    const float* __restrict__ xs, const float* __restrict__ dts,
    const float* __restrict__ As, const float* __restrict__ Bs,
    const float* __restrict__ Cs, const float* __restrict__ D,
    float* __restrict__ out) {
  // Double-buffered chunk storage: 2 bufs x 5 arrays x TC x CPB floats = 160 KB
  __shared__ float smem[2 * 5 * TC * CPB];

  const int lane = threadIdx.x;                  // 0..31
  const int b    = blockIdx.x >> 5;              // / (DMSZ/CPB)
  const int d0   = (blockIdx.x & 31) * CPB;

  const size_t chanoff = (size_t)b * LSZ * DMSZ + (size_t)d0;
  const float* base0 = xs  + chanoff;
  const float* base1 = dts + chanoff;
  const float* base2 = As  + chanoff;
  const float* base3 = Bs  + chanoff;
  const float* base4 = Cs  + chanoff;

  // Flat-address low 32 bits of an LDS pointer == wave-relative LDS byte offset
  const uint32_t lds0 = (uint32_t)(uintptr_t)(&smem[0]);
  const uint32_t ABYTES = (uint32_t)(TC * CPB * 4);  // one array-plane in bytes

  // Prologue: stream chunk 0 into buffer 0
  {
    uint32_t la = lds0;
    tdm_load_tile(la + 0 * ABYTES, base0);
    tdm_load_tile(la + 1 * ABYTES, base1);
    tdm_load_tile(la + 2 * ABYTES, base2);
    tdm_load_tile(la + 3 * ABYTES, base3);
    tdm_load_tile(la + 4 * ABYTES, base4);
  }

  float state = 0.0f;
  const float Dd = D[d0 + lane];
  float* optr = out + chanoff + lane;

  for (int c = 0; c < NCHUNK; ++c) {
    const int buf = c & 1;
    if (c + 1 < NCHUNK) {
      // Prefetch next chunk into the other buffer, then wait until only the
      // 5 prefetch loads remain outstanding (TDM completes in order per wave)
      const size_t off = (size_t)(c + 1) * TC * DMSZ;
      const uint32_t la = lds0 + (uint32_t)(buf ^ 1) * 5u * ABYTES;
      tdm_load_tile(la + 0 * ABYTES, base0 + off);
      tdm_load_tile(la + 1 * ABYTES, base1 + off);
      tdm_load_tile(la + 2 * ABYTES, base2 + off);
      tdm_load_tile(la + 3 * ABYTES, base3 + off);
      tdm_load_tile(la + 4 * ABYTES, base4 + off);
      __builtin_amdgcn_s_wait_tensorcnt((short)5);
    } else {
      __builtin_amdgcn_s_wait_tensorcnt((short)0);
    }

    const float* p  = &smem[(size_t)buf * 5 * TC * CPB];
    const float* px = p;
    const float* pd = p + 1 * TC * CPB;
    const float* pa = p + 2 * TC * CPB;
    const float* pb = p + 3 * TC * CPB;
    const float* pc = p + 4 * TC * CPB;
    float* o = optr + (size_t)c * TC * DMSZ;

#pragma unroll 4
    for (int t = 0; t < TC; ++t) {
      const int idx = t * CPB + lane;
      const float x   = px[idx];
      const float dtr = pd[idx];
      const float a   = pa[idx];
      const float bb  = pb[idx];
      const float cc  = pc[idx];

      const float sp = softplus_fast(dtr);        // off the critical path
      const float z  = fmaf(a, state, bb * x);    // chain: fma
      const float th = tanh_fast(z);              // chain: tanh (TRANS)
      state = fmaf(sp, th, state);                // chain: fma
      const float y = fmaf(Dd, x, cc * state);
      __builtin_nontemporal_store(y, o);
      o += DMSZ;
    }
  }
}

extern "C" void kernel_launch(void* const* d_in, const int* in_sizes, int n_in,
                              void* d_out, int out_size, void* d_ws, size_t ws_size,
                              hipStream_t stream) {
  const float* xs  = (const float*)d_in[0];
  const float* dts = (const float*)d_in[1];
  const float* As  = (const float*)d_in[2];
  const float* Bs  = (const float*)d_in[3];
  const float* Cs  = (const float*)d_in[4];
  const float* D   = (const float*)d_in[5];
  float* out = (float*)d_out;

  dim3 grid(BSZ * (DMSZ / CPB));  // 128 blocks, one wave32 each
  dim3 block(CPB);                // 32 threads
  SS2D_scan_kernel<<<grid, block, 0, stream>>>(xs, dts, As, Bs, Cs, D, out);
}